// MyCNN_50955492000195
// MI455X (gfx1250) — compile-verified
//
#include <hip/hip_runtime.h>

typedef __attribute__((ext_vector_type(2))) float v2f;
typedef __attribute__((ext_vector_type(8))) float v8f;

#define CH 8
#define NB 16384
#define CONV_HW 196              /* 14*14 */
#define NPOS (NB * CONV_HW)      /* 3,211,264 conv outputs per channel */
#define STAT_BLOCKS 1024
#define PSTRIDE 393              /* 392 + 1 pad: odd stride -> conflict-free LDS */

// ---------------------------------------------------------------------------
// Kernel 1: conv (recomputed, not stored) -> per-channel sum / sumsq partials.
// Deterministic: wave shuffle reduce -> LDS -> per-block partial in workspace.
// ---------------------------------------------------------------------------
__global__ __launch_bounds__(256) void k_conv_stats(
    const float* __restrict__ x,
    const float* __restrict__ conv_w,
    const float* __restrict__ conv_b,
    float* __restrict__ partial)   // [STAT_BLOCKS][16]: 8 sums then 8 sumsqs
{
    __shared__ float s_w[72];
    __shared__ float s_b[8];
    __shared__ float s_part[8][16];

    const int t = threadIdx.x;
    if (t < 72) s_w[t] = conv_w[t];
    if (t < 8)  s_b[t] = conv_b[t];
    __syncthreads();

    float s[CH], ss[CH];
#pragma unroll
    for (int c = 0; c < CH; ++c) { s[c] = 0.0f; ss[c] = 0.0f; }

    for (int idx = blockIdx.x * 256 + t; idx < NPOS; idx += gridDim.x * 256) {
        const int b  = idx / CONV_HW;
        const int p  = idx - b * CONV_HW;
        const int oh = p / 14;
        const int ow = p - oh * 14;
        const float* xb = x + (size_t)b * 784;

        float patch[9];
        const int ih0 = 2 * oh - 1, iw0 = 2 * ow - 1;
#pragma unroll
        for (int i = 0; i < 3; ++i)
#pragma unroll
            for (int j = 0; j < 3; ++j) {
                const int ih = ih0 + i, iw = iw0 + j;
                const bool ok = (ih >= 0) & (ih < 28) & (iw >= 0) & (iw < 28);
                patch[i * 3 + j] = ok ? xb[ih * 28 + iw] : 0.0f;
            }
#pragma unroll
        for (int c = 0; c < CH; ++c) {
            float v = s_b[c];
#pragma unroll
            for (int k = 0; k < 9; ++k) v += s_w[c * 9 + k] * patch[k];
            s[c]  += v;
            ss[c] += v * v;
        }
    }

    // wave32 tree reduce
#pragma unroll
    for (int c = 0; c < CH; ++c) {
        for (int off = 16; off > 0; off >>= 1) {
            s[c]  += __shfl_down(s[c],  off, 32);
            ss[c] += __shfl_down(ss[c], off, 32);
        }
    }
    const int wave = t >> 5, lane = t & 31;
    if (lane == 0) {
#pragma unroll
        for (int c = 0; c < CH; ++c) {
            s_part[wave][c]     = s[c];
            s_part[wave][8 + c] = ss[c];
        }
    }
    __syncthreads();
    if (t < 16) {
        float acc = 0.0f;
        for (int w = 0; w < 8; ++w) acc += s_part[w][t];   // fixed order
        partial[blockIdx.x * 16 + t] = acc;
    }
}

// ---------------------------------------------------------------------------
// Kernel 2: fold batch stats + gamma/beta + conv bias into per-channel
// scale / shift.  Single block, fixed-order serial reductions => deterministic.
// ---------------------------------------------------------------------------
__global__ __launch_bounds__(32) void k_bn_params(
    const float* __restrict__ partial,
    const float* __restrict__ gamma,
    const float* __restrict__ beta,
    float* __restrict__ bn)        // [0..7]=scale, [8..15]=shift (pre conv-bias)
{
    __shared__ float stats[16];
    const int t = threadIdx.x;
    if (t < 16) {
        float acc = 0.0f;
        for (int bl = 0; bl < STAT_BLOCKS; ++bl) acc += partial[bl * 16 + t];
        stats[t] = acc;
    }
    __syncthreads();
    if (t < 8) {
        const float invN = 1.0f / (float)NPOS;
        const float mean = stats[t] * invN;
        const float var  = stats[8 + t] * invN - mean * mean;
        const float sc   = gamma[t] * rsqrtf(var + 1e-5f);
        bn[t]     = sc;
        bn[8 + t] = beta[t] - mean * sc;
    }
}

// ---------------------------------------------------------------------------
// Kernel 3: per 16-row tile: recompute conv -> BN -> ReLU -> maxpool into LDS,
// then FC via V_WMMA_F32_16X16X4_F32 (wave 0), D-tile stored to out[16][10].
// ---------------------------------------------------------------------------
__global__ __launch_bounds__(128) void k_fused(
    const float* __restrict__ x,
    const float* __restrict__ conv_w,
    const float* __restrict__ conv_b,
    const float* __restrict__ fc_w,    // [10][392]
    const float* __restrict__ fc_b,    // [10]
    const float* __restrict__ bn,      // scale[8], shift[8]
    float* __restrict__ out)           // [NB][10]
{
    __shared__ float s_pool[16 * PSTRIDE];   // pooled activations, A-matrix src
    __shared__ float s_bw[392 * 16];         // fc_w^T zero-padded to 16 cols
    __shared__ float s_w[72];
    __shared__ float s_a[8];                 // BN scale
    __shared__ float s_d[8];                 // scale*conv_b + shift

    const int t  = threadIdx.x;
    const int b0 = blockIdx.x * 16;

    if (t < 72) s_w[t] = conv_w[t];
    if (t >= 96 && t < 104) {
        const int c = t - 96;
        const float sc = bn[c];
        s_a[c] = sc;
        s_d[c] = sc * conv_b[c] + bn[8 + c];
    }
    for (int i = t; i < 392 * 16; i += 128) {
        const int n = i & 15, k = i >> 4;
        s_bw[i] = (n < 10) ? fc_w[n * 392 + k] : 0.0f;
    }
    __syncthreads();

    // ---- Phase 1: conv+BN+ReLU+maxpool for 16 batch rows -> LDS ----
    for (int idx = t; idx < 16 * 49; idx += 128) {
        const int bi = idx / 49;
        const int p  = idx - bi * 49;
        const int ph = p / 7, pw = p - ph * 7;
        const float* xb = x + (size_t)(b0 + bi) * 784;

        float patch[5][5];
        const int r0 = 4 * ph - 1, c0 = 4 * pw - 1;
#pragma unroll
        for (int i = 0; i < 5; ++i)
#pragma unroll
            for (int j = 0; j < 5; ++j) {
                const int r = r0 + i, cc = c0 + j;
                const bool ok = (r >= 0) & (r < 28) & (cc >= 0) & (cc < 28);
                patch[i][j] = ok ? xb[r * 28 + cc] : 0.0f;
            }
#pragma unroll
        for (int c = 0; c < CH; ++c) {
            float best = 0.0f;   // ReLU outputs are >= 0, so 0 is a valid floor
#pragma unroll
            for (int dy = 0; dy < 2; ++dy)
#pragma unroll
                for (int dx = 0; dx < 2; ++dx) {
                    float v = 0.0f;
#pragma unroll
                    for (int i = 0; i < 3; ++i)
#pragma unroll
                        for (int j = 0; j < 3; ++j)
                            v += s_w[c * 9 + i * 3 + j] *
                                 patch[2 * dy + i][2 * dx + j];
                    const float z = s_a[c] * v + s_d[c];
                    best = fmaxf(best, z);
                }
            s_pool[bi * PSTRIDE + c * 49 + p] = best;
        }
    }
    __syncthreads();

    // ---- Phase 2: FC as 98 x V_WMMA_F32_16X16X4_F32 on wave 0 ----
    if (t < 32) {                       // wave-uniform: EXEC all-1 for WMMA
        const int lane = t;
        const int m  = lane & 15;            // A row
        const int n  = lane & 15;            // B/D column
        const int kb = (lane >> 4) << 1;     // K offset 0 or 2 for this half

        v8f acc;
        const float bias = (n < 10) ? fc_b[n] : 0.0f;
#pragma unroll
        for (int r = 0; r < 8; ++r) acc[r] = bias;

        for (int k0 = 0; k0 < 392; k0 += 4) {
            v2f a, b;
            a.x = s_pool[m * PSTRIDE + k0 + kb];
            a.y = s_pool[m * PSTRIDE + k0 + kb + 1];
            b.x = s_bw[(k0 + kb)     * 16 + n];
            b.y = s_bw[(k0 + kb + 1) * 16 + n];
            acc = __builtin_amdgcn_wmma_f32_16x16x4_f32(
                false, a, false, b, (short)0, acc, false, false);
        }

        if (n < 10) {
            const int rowbase = b0 + ((lane >> 4) << 3);  // M = r + 8*(lane>=16)
#pragma unroll
            for (int r = 0; r < 8; ++r)
                out[(size_t)(rowbase + r) * 10 + n] = acc[r];
        }
    }
}

// ---------------------------------------------------------------------------
extern "C" void kernel_launch(void* const* d_in, const int* in_sizes, int n_in,
                              void* d_out, int out_size, void* d_ws, size_t ws_size,
                              hipStream_t stream) {
    const float* x      = (const float*)d_in[0];
    const float* conv_w = (const float*)d_in[1];
    const float* conv_b = (const float*)d_in[2];
    const float* gamma  = (const float*)d_in[3];
    const float* beta   = (const float*)d_in[4];
    const float* fc_w   = (const float*)d_in[5];
    const float* fc_b   = (const float*)d_in[6];
    float* out = (float*)d_out;

    float* partial = (float*)d_ws;                 // STAT_BLOCKS*16 floats
    float* bn      = partial + STAT_BLOCKS * 16;   // 16 floats

    k_conv_stats<<<STAT_BLOCKS, 256, 0, stream>>>(x, conv_w, conv_b, partial);
    k_bn_params<<<1, 32, 0, stream>>>(partial, gamma, beta, bn);
    k_fused<<<NB / 16, 128, 0, stream>>>(x, conv_w, conv_b, fc_w, fc_b, bn, out);
}